// DynamicGraphConstructor_6640019440024
// MI455X (gfx1250) — compile-verified
//
#include <hip/hip_runtime.h>
#include <hip/hip_bf16.h>
#include <math.h>

typedef float v2f __attribute__((ext_vector_type(2)));
typedef float v8f __attribute__((ext_vector_type(8)));

typedef __attribute__((address_space(1))) int gint_as;   // global (addrspace 1)
typedef __attribute__((address_space(3))) int lint_as;   // LDS (addrspace 3)

#if defined(__has_builtin)
#if __has_builtin(__builtin_amdgcn_global_load_async_to_lds_b32) && \
    __has_builtin(__builtin_amdgcn_s_wait_asynccnt)
#define HAVE_ASYNC 1
#endif
#endif
#ifndef HAVE_ASYNC
#define HAVE_ASYNC 0
#endif

#define BB 8
#define LL 12
#define NN 1024
#define HH 32
#define NDD 16
#define ROWS (BB * NN)          // 8192
#define H2 (2 * HH)             // 64
#define BN_EPS 1e-5f

// workspace layout (float offsets)
#define OFF_H      0                        // 8192*64
#define OFF_SCALE  (OFF_H + ROWS * H2)      // 64
#define OFF_SHIFT  (OFF_SCALE + 64)         // 64
#define OFF_DY     (OFF_SHIFT + 64)         // 8192*32
#define OFF_QK     (OFF_DY + ROWS * HH)     // 4 * 8192*32
#define OFF_G      (OFF_QK + 4 * ROWS * HH) // 2*8*1024*1024

// ---------------- K1: h = relu(X @ W1 + b1), X built from history_data ----------------
__global__ void k_hidden(const float* __restrict__ hist, const float* __restrict__ W1,
                         const float* __restrict__ b1, float* __restrict__ h) {
    int idx = blockIdx.x * blockDim.x + threadIdx.x;   // ROWS*H2 threads
    int row = idx >> 6;      // /64
    int c   = idx & 63;
    int b   = row >> 10;
    int n   = row & (NN - 1);
    float acc = b1[c];
    #pragma unroll
    for (int l = 0; l < LL; ++l)
        acc = fmaf(hist[b * (LL * NN) + l * NN + n], W1[l * H2 + c], acc);
    h[idx] = fmaxf(acc, 0.0f);
}

// ---------------- K2: per-column batchnorm stats -> scale/shift ----------------
__global__ void k_stats(const float* __restrict__ h, const float* __restrict__ gamma,
                        const float* __restrict__ beta, float* __restrict__ scale,
                        float* __restrict__ shift) {
    __shared__ float rs[256], rs2[256];
    int j = blockIdx.x;          // 0..63
    int tid = threadIdx.x;
    float s = 0.0f, s2 = 0.0f;
    for (int r = tid; r < ROWS; r += 256) {
        float v = h[r * H2 + j];
        s += v; s2 = fmaf(v, v, s2);
    }
    rs[tid] = s; rs2[tid] = s2;
    __syncthreads();
    for (int st = 128; st > 0; st >>= 1) {
        if (tid < st) { rs[tid] += rs[tid + st]; rs2[tid] += rs2[tid + st]; }
        __syncthreads();
    }
    if (tid == 0) {
        float mu  = rs[0] * (1.0f / ROWS);
        float var = rs2[0] * (1.0f / ROWS) - mu * mu;
        float sc  = gamma[j] * rsqrtf(var + BN_EPS);
        scale[j] = sc;
        shift[j] = beta[j] - mu * sc;
    }
}

// ---------------- K3a: dy = (h*scale+shift) @ W2 + b2 ----------------
__global__ void k_dy(const float* __restrict__ h, const float* __restrict__ scale,
                     const float* __restrict__ shift, const float* __restrict__ W2,
                     const float* __restrict__ b2, float* __restrict__ dy) {
    int idx = blockIdx.x * blockDim.x + threadIdx.x;  // ROWS*HH threads
    int row = idx >> 5;
    int k   = idx & 31;
    float acc = b2[k];
    const float* hr = h + row * H2;
    #pragma unroll 8
    for (int j = 0; j < H2; ++j)
        acc = fmaf(fmaf(hr[j], scale[j], shift[j]), W2[j * HH + k], acc);
    dy[idx] = acc;
}

// ---------------- K3b: Q/K for both embeddings ----------------
// qk layout: [a][row][h], a = e*2 + t (t: 0=Q, 1=K)
__global__ void k_qk(const float* __restrict__ dy, const float* __restrict__ embd,
                     const float* __restrict__ embu, const float* __restrict__ WQ,
                     const float* __restrict__ WK, float* __restrict__ qk) {
    int idx = blockIdx.x * blockDim.x + threadIdx.x;  // 4*ROWS*HH threads
    int hcol = idx & 31;
    int row  = (idx >> 5) & (ROWS - 1);
    int a    = idx >> 18;              // 0..3
    int e = a >> 1, t = a & 1;
    const float* Wm  = t ? WK : WQ;
    const float* emb = e ? embu : embd;
    int n = row & (NN - 1);
    float acc = 0.0f;
    const float* dr = dy + row * HH;
    #pragma unroll 8
    for (int j = 0; j < HH; ++j)
        acc = fmaf(dr[j], Wm[j * HH + hcol], acc);
    const float* er = emb + n * NDD;
    #pragma unroll
    for (int j = 0; j < NDD; ++j)
        acc = fmaf(er[j], Wm[(HH + j) * HH + hcol], acc);
    qk[a * (ROWS * HH) + row * HH + hcol] = acc;
}

// ---------------- K4: logits -> softmax -> mask -> row-normalize ----------------
__global__ void k_attn(const float* __restrict__ qk, const float* __restrict__ adj0,
                       const float* __restrict__ adj1, float* __restrict__ g,
                       float* __restrict__ out) {
    __shared__ float qrow[HH];
    __shared__ float lg[NN];
    __shared__ float red[256];
    int n = blockIdx.x, b = blockIdx.y, e = blockIdx.z;
    int tid = threadIdx.x;
    int row = b * NN + n;
    const float* Qb = qk + (e * 2 + 0) * (ROWS * HH);
    const float* Kb = qk + (e * 2 + 1) * (ROWS * HH);
    if (tid < HH) qrow[tid] = Qb[row * HH + tid];
    __syncthreads();

    const float sc = 0.17677669529663687f;   // 1/sqrt(32)
    for (int m = tid; m < NN; m += 256) {
        const float* kr = Kb + (b * NN + m) * HH;
        float acc = 0.0f;
        #pragma unroll 8
        for (int hcol = 0; hcol < HH; ++hcol)
            acc = fmaf(qrow[hcol], kr[hcol], acc);
        lg[m] = acc * sc;
    }
    __syncthreads();

    float lmax = -INFINITY;
    for (int m = tid; m < NN; m += 256) lmax = fmaxf(lmax, lg[m]);
    red[tid] = lmax; __syncthreads();
    for (int st = 128; st > 0; st >>= 1) {
        if (tid < st) red[tid] = fmaxf(red[tid], red[tid + st]);
        __syncthreads();
    }
    float vmax = red[0];
    __syncthreads();

    float lsum = 0.0f;
    for (int m = tid; m < NN; m += 256) {
        float v = __expf(lg[m] - vmax);
        lg[m] = v; lsum += v;
    }
    red[tid] = lsum; __syncthreads();
    for (int st = 128; st > 0; st >>= 1) {
        if (tid < st) red[tid] += red[tid + st];
        __syncthreads();
    }
    float inv_soft = 1.0f / red[0];
    __syncthreads();

    const float* adj = e ? adj1 : adj0;
    float msum = 0.0f;
    for (int m = tid; m < NN; m += 256) {
        float w = lg[m] * inv_soft * (adj[n * NN + m] + 1e-7f);
        lg[m] = w; msum += w;
    }
    red[tid] = msum; __syncthreads();
    for (int st = 128; st > 0; st >>= 1) {
        if (tid < st) red[tid] += red[tid + st];
        __syncthreads();
    }
    float inv = 1.0f / red[0];
    if (!isfinite(inv)) inv = 0.0f;

    float* grow = g + ((size_t)(e * BB + b) * NN + n) * NN;
    float* orow = out + ((((size_t)(2 * e) * BB + b) * NN + n) * (2 * NN));
    for (int m = tid; m < NN; m += 256) {
        float gv = lg[m] * inv;
        grow[m] = gv;
        float ov = (m == n) ? 0.0f : gv;
        orow[m] = ov;
        orow[m + NN] = ov;                 // K_T = 2 row duplication
    }
}

// ---------------- K5: S = g @ g, LDS-staged B panel, double-buffered async fills ----------
// grid (64, 16), 256 threads = 8 waves; block computes a 128x128 tile.
// B staged K-transposed in LDS: Blds[buf][col*18 + k], stride 18 => 8B-aligned b64
// fragment reads, conflict-free banks.
#define KC 16
#define BSTRIDE 18

__device__ __forceinline__ void fill_chunk(const float* __restrict__ G, int k0, int col0,
                                           float* __restrict__ dst, int tid) {
    #pragma unroll
    for (int i = 0; i < 8; ++i) {
        int q  = tid + i * 256;           // 2048 elements = 16 x 128 chunk
        int c  = q & 127;
        int kk = q >> 7;
        const float* gp = G + (size_t)(k0 + kk) * NN + col0 + c;
        float* lp = dst + c * BSTRIDE + kk;
#if HAVE_ASYNC
        __builtin_amdgcn_global_load_async_to_lds_b32(
            (gint_as*)(size_t)gp, (lint_as*)(unsigned)(size_t)lp, 0, 0);
#else
        *lp = *gp;
#endif
    }
}

__global__ void __launch_bounds__(256) k_gemm2(const float* __restrict__ g,
                                               float* __restrict__ out) {
    __shared__ __align__(16) float Blds[2][128 * BSTRIDE];

    int mat  = blockIdx.y;              // e*8 + b
    int e    = mat >> 3;
    int b    = mat & 7;
    int tid  = threadIdx.x;
    int lane = tid & 31;
    int wave = tid >> 5;                // 0..7
    int tileM = blockIdx.x >> 3;        // 128-row block (0..7)
    int tileN = blockIdx.x & 7;         // 128-col block (0..7)

    const float* G = g + (size_t)mat * NN * NN;
    int half = lane >> 4;
    int l15  = lane & 15;
    int rowA = tileM * 128 + wave * 16 + l15;
    int col0 = tileN * 128;

    v8f acc[8];
    #pragma unroll
    for (int i = 0; i < 8; ++i) acc[i] = (v8f){};

    const float* Arow = G + (size_t)rowA * NN;

    fill_chunk(G, 0, col0, &Blds[0][0], tid);

    for (int kc = 0; kc < NN / KC; ++kc) {
        int cur = kc & 1;
#if HAVE_ASYNC
        __builtin_amdgcn_s_wait_asynccnt(0);
#endif
        __syncthreads();
        if (kc + 1 < NN / KC)
            fill_chunk(G, (kc + 1) * KC, col0, &Blds[cur ^ 1][0], tid);

        int k0 = kc * KC;
        const float* Bbuf = &Blds[cur][0];
        #pragma unroll
        for (int ks = 0; ks < 4; ++ks) {
            int kr = ks * 4 + 2 * half;       // even -> 8B aligned
            v2f a = *(const v2f*)(Arow + k0 + kr);
            // load all 8 B fragments into distinct registers first so the
            // scheduler can clause the ds loads and stagger s_wait_dscnt
            v2f bf[8];
            #pragma unroll
            for (int nb = 0; nb < 8; ++nb)
                bf[nb] = *(const v2f*)(Bbuf + (nb * 16 + l15) * BSTRIDE + kr);
            #pragma unroll
            for (int nb = 0; nb < 8; ++nb)
                acc[nb] = __builtin_amdgcn_wmma_f32_16x16x4_f32(
                    false, a, false, bf[nb], (short)0, acc[nb], false, false);
        }
    }

    // store: offdiag mask + K_T row duplication; D layout: VGPR r -> M=r+8*half, N=l15
    float* obase = out + (((size_t)(2 * e + 1) * BB + b) * NN) * (2 * NN);
    #pragma unroll
    for (int r = 0; r < 8; ++r) {
        int rowC = tileM * 128 + wave * 16 + r + 8 * half;
        float* orow = obase + (size_t)rowC * (2 * NN);
        #pragma unroll
        for (int nb = 0; nb < 8; ++nb) {
            int colC = tileN * 128 + nb * 16 + l15;
            float v = acc[nb][r];
            if (rowC == colC) v = 0.0f;
            orow[colC] = v;
            orow[colC + NN] = v;
        }
    }
}

extern "C" void kernel_launch(void* const* d_in, const int* in_sizes, int n_in,
                              void* d_out, int out_size, void* d_ws, size_t ws_size,
                              hipStream_t stream) {
    const float* hist  = (const float*)d_in[0];
    const float* embd  = (const float*)d_in[1];
    const float* embu  = (const float*)d_in[2];
    const float* adj0  = (const float*)d_in[3];
    const float* adj1  = (const float*)d_in[4];
    const float* W1    = (const float*)d_in[5];
    const float* b1    = (const float*)d_in[6];
    const float* gamma = (const float*)d_in[7];
    const float* beta  = (const float*)d_in[8];
    const float* W2    = (const float*)d_in[9];
    const float* b2    = (const float*)d_in[10];
    const float* WQ    = (const float*)d_in[11];
    const float* WK    = (const float*)d_in[12];
    float* out = (float*)d_out;
    float* ws  = (float*)d_ws;

    float* ws_h     = ws + OFF_H;
    float* ws_scale = ws + OFF_SCALE;
    float* ws_shift = ws + OFF_SHIFT;
    float* ws_dy    = ws + OFF_DY;
    float* ws_qk    = ws + OFF_QK;
    float* ws_g     = ws + OFF_G;

    k_hidden<<<(ROWS * H2) / 256, 256, 0, stream>>>(hist, W1, b1, ws_h);
    k_stats<<<H2, 256, 0, stream>>>(ws_h, gamma, beta, ws_scale, ws_shift);
    k_dy<<<(ROWS * HH) / 256, 256, 0, stream>>>(ws_h, ws_scale, ws_shift, W2, b2, ws_dy);
    k_qk<<<(4 * ROWS * HH) / 256, 256, 0, stream>>>(ws_dy, embd, embu, WQ, WK, ws_qk);
    k_attn<<<dim3(NN, BB, 2), 256, 0, stream>>>(ws_qk, adj0, adj1, ws_g, out);
    k_gemm2<<<dim3(64, 16), 256, 0, stream>>>(ws_g, out);
}